// CausalSelfAttention_TSSA_15401752723500
// MI455X (gfx1250) — compile-verified
//
#include <hip/hip_runtime.h>
#include <hip/hip_bf16.h>

typedef float v2f __attribute__((ext_vector_type(2)));
typedef float v8f __attribute__((ext_vector_type(8)));

#define BATCH 8
#define NSEQ  2048
#define CH    2048
#define NH    16
#define HD    128
#define EPS_F16 0.0009765625f   // fp16 eps, per reference

// ---------------------------------------------------------------------------
// GEMM: OUT[m, o] = sum_k X[m, k] * W[o, k]   (einsum 'bnc,oc->bno')
// M = BATCH*NSEQ = 16384, O = K = CH = 2048.
// Each wave computes a 16x64 strip: 4 accumulators share one A fragment per
// V_WMMA_F32_16X16X4_F32 step (5 loads / 4 WMMAs instead of 8 / 4).
// Consecutive waves in a block take consecutive tile_m with the same o-group,
// so the 64 W rows stay resident in L0/L2.
// HEAD_LAYOUT=true  -> write w in (B, H, N, D) layout (for the scans)
// HEAD_LAYOUT=false -> write row-major (M, CH) (final output)
// ---------------------------------------------------------------------------
template <bool HEAD_LAYOUT>
__global__ __launch_bounds__(256)
void wmma_gemm_xWt(const float* __restrict__ X, const float* __restrict__ W,
                   float* __restrict__ OUT)
{
    const int lane = threadIdx.x & 31;
    const int wave = threadIdx.x >> 5;
    const int gw   = blockIdx.x * 8 + wave;

    const int tiles_m = (BATCH * NSEQ) / 16;   // 1024
    const int tile_m  = gw % tiles_m;
    const int tile_og = gw / tiles_m;          // 0..31: group of 4 o-tiles
    const int m0 = tile_m * 16;
    const int o0 = tile_og * 64;

    const int lo = lane & 15;                  // row/col within tile
    const int hi = lane >> 4;                  // K sub-slot (A/B: K + 2*hi)

    // A fragment: lane -> row (m0+lo), K pair starting at (k + 2*hi)
    const float* arow = X + (size_t)(m0 + lo) * CH + 2 * hi;
    // B fragments: lane -> col (o0 + 16*j + lo), same K pair (W is (O,K) row-major)
    const float* brow = W + (size_t)(o0 + lo) * CH + 2 * hi;

    v8f acc[4];
#pragma unroll
    for (int j = 0; j < 4; ++j) acc[j] = (v8f){0.f,0.f,0.f,0.f,0.f,0.f,0.f,0.f};

    for (int k = 0; k < CH; k += 16) {
#pragma unroll
        for (int kk = 0; kk < 16; kk += 4) {
            const v2f a  = *(const v2f*)(arow + k + kk);
            const v2f b0 = *(const v2f*)(brow + (size_t) 0 * CH + k + kk);
            const v2f b1 = *(const v2f*)(brow + (size_t)16 * CH + k + kk);
            const v2f b2 = *(const v2f*)(brow + (size_t)32 * CH + k + kk);
            const v2f b3 = *(const v2f*)(brow + (size_t)48 * CH + k + kk);
            acc[0] = __builtin_amdgcn_wmma_f32_16x16x4_f32(
                false, a, false, b0, (short)0, acc[0], false, false);
            acc[1] = __builtin_amdgcn_wmma_f32_16x16x4_f32(
                false, a, false, b1, (short)0, acc[1], false, false);
            acc[2] = __builtin_amdgcn_wmma_f32_16x16x4_f32(
                false, a, false, b2, (short)0, acc[2], false, false);
            acc[3] = __builtin_amdgcn_wmma_f32_16x16x4_f32(
                false, a, false, b3, (short)0, acc[3], false, false);
        }
    }

    // C/D layout: VGPR r -> M = r + 8*hi, N = lane&15
    const int mbase = m0 + hi * 8;
#pragma unroll
    for (int j = 0; j < 4; ++j) {
        const int ocol = o0 + 16 * j + lo;
#pragma unroll
        for (int r = 0; r < 8; ++r) {
            const int m = mbase + r;
            const float v = acc[j][r];
            if (HEAD_LAYOUT) {
                const int b = m >> 11;                 // / NSEQ
                const int n = m & (NSEQ - 1);
                const int h = ocol >> 7;               // / HD
                const int d = ocol & (HD - 1);
                OUT[(((size_t)(b * NH + h)) * NSEQ + n) * HD + d] = v;
            } else {
                OUT[(size_t)m * CH + ocol] = v;
            }
        }
    }
}

// ---------------------------------------------------------------------------
// Scan 1 + head logits:
//   denom = max(cumsum_n(w^2), eps); contrib = w^2/denom
//   tmp[b,h,n] = (sum_d contrib + D * denom_bias[h,n]) * temp[h]
// One block per (b,h); thread t owns feature d=t and carries a private
// running cumsum along n. Cross-d reduction goes through a 64x128 LDS tile
// with rotated column access (conflict-free on 64 banks).
// ---------------------------------------------------------------------------
__global__ __launch_bounds__(128)
void scan1_tmp(const float* __restrict__ Wbuf, const float* __restrict__ temp,
               const float* __restrict__ dbias, float* __restrict__ tmp_out)
{
    __shared__ float tile[64 * 128];      // 32 KB
    __shared__ float partial[128];

    const int bh = blockIdx.x;            // b*NH + h
    const int h  = bh & (NH - 1);
    const int t  = threadIdx.x;           // = d
    const float tscale = temp[h];

    const float* wp = Wbuf + (size_t)bh * NSEQ * HD + t;
    float cum = 0.f;

    for (int chunk = 0; chunk < NSEQ / 64; ++chunk) {
        const int nbase = chunk * 64;
#pragma unroll 4
        for (int i = 0; i < 64; ++i) {
            const float wv  = wp[(size_t)(nbase + i) * HD];
            const float wsq = wv * wv;
            cum += wsq;
            tile[i * 128 + t] = wsq / fmaxf(cum, EPS_F16);
        }
        __syncthreads();

        // threads t<64 and t>=64 each sum one half of row (t & 63)
        const int row  = t & 63;
        const int half = (t >> 6) * 64;
        float s = 0.f;
#pragma unroll 8
        for (int j = 0; j < 64; ++j) {
            const int col = half + ((j + t) & 63);   // rotate: conflict-free
            s += tile[row * 128 + col];
        }
        partial[t] = s;
        __syncthreads();

        if (t < 64) {
            const int n = nbase + t;
            const float stot = partial[t] + partial[t + 64];
            tmp_out[(size_t)bh * NSEQ + n] =
                (stot + (float)HD * dbias[h * NSEQ + n]) * tscale;
        }
        __syncthreads();
    }
}

// ---------------------------------------------------------------------------
// Softmax across the 16 heads: Pi[b,h,n] = softmax_h(tmp[b,h,n])
// One thread per (b,n); 16 values in registers.
// ---------------------------------------------------------------------------
__global__ __launch_bounds__(256)
void head_softmax(const float* __restrict__ tmp_in, float* __restrict__ Pi)
{
    const int idx = blockIdx.x * 256 + threadIdx.x;   // b*NSEQ + n
    if (idx >= BATCH * NSEQ) return;
    const int b = idx >> 11;
    const int n = idx & (NSEQ - 1);

    float v[NH];
    float mx = -INFINITY;
#pragma unroll
    for (int h = 0; h < NH; ++h) {
        v[h] = tmp_in[((size_t)(b * NH + h)) * NSEQ + n];
        mx = fmaxf(mx, v[h]);
    }
    float sum = 0.f;
#pragma unroll
    for (int h = 0; h < NH; ++h) {
        v[h] = __expf(v[h] - mx);
        sum += v[h];
    }
    const float inv = 1.f / sum;
#pragma unroll
    for (int h = 0; h < NH; ++h)
        Pi[((size_t)(b * NH + h)) * NSEQ + n] = v[h] * inv;
}

// ---------------------------------------------------------------------------
// Scan 2 + y:
//   dots = cumsum_n(w^2 * Pi) / (cumsum_n(Pi) + eps)
//   y = -(w * Pi) / (1 + dots), written in (B, N, C) layout for GEMM2.
// One block per (b,h); Pi row staged in LDS; thread t owns d=t.
// ---------------------------------------------------------------------------
__global__ __launch_bounds__(128)
void scan2_y(const float* __restrict__ Wbuf, const float* __restrict__ Pi,
             float* __restrict__ Y)
{
    __shared__ float pil[NSEQ];           // 8 KB

    const int bh = blockIdx.x;
    const int b  = bh >> 4;
    const int h  = bh & (NH - 1);
    const int t  = threadIdx.x;           // = d

    for (int j = t; j < NSEQ; j += 128)
        pil[j] = Pi[(size_t)bh * NSEQ + j];
    __syncthreads();

    const float* wp = Wbuf + (size_t)bh * NSEQ * HD + t;
    float* yp = Y + (size_t)b * NSEQ * CH + h * HD + t;

    float cumP = 0.f, cumWP = 0.f;
    for (int n = 0; n < NSEQ; ++n) {
        const float p  = pil[n];
        const float wv = wp[(size_t)n * HD];
        cumP  += p;
        cumWP += wv * wv * p;
        const float dots = cumWP / (cumP + EPS_F16);
        const float attn = 1.0f / (1.0f + dots);
        yp[(size_t)n * CH] = -(wv * p) * attn;
    }
}

// ---------------------------------------------------------------------------
// Launch. Workspace layout (floats):
//   y   : BATCH*NSEQ*CH          (134 MB)  — GEMM2 input
//   tmp : BATCH*NH*NSEQ          (1 MB)
//   Pi  : BATCH*NH*NSEQ          (1 MB)
// w (B,H,N,D) is stashed in d_out — it is dead before GEMM2 overwrites d_out.
// ---------------------------------------------------------------------------
extern "C" void kernel_launch(void* const* d_in, const int* in_sizes, int n_in,
                              void* d_out, int out_size, void* d_ws, size_t ws_size,
                              hipStream_t stream)
{
    (void)in_sizes; (void)n_in; (void)out_size; (void)ws_size;

    const float* x      = (const float*)d_in[0];
    const float* W_attn = (const float*)d_in[1];
    const float* W_proj = (const float*)d_in[2];
    const float* temp   = (const float*)d_in[3];
    const float* dbias  = (const float*)d_in[4];

    float* out  = (float*)d_out;
    float* ws   = (float*)d_ws;
    float* y    = ws;
    float* tmpb = ws + (size_t)BATCH * NSEQ * CH;
    float* Pi   = tmpb + (size_t)BATCH * NH * NSEQ;
    float* wbuf = out;   // stage w in d_out

    // 1024 m-tiles * 32 o-groups = 32768 waves / 8 waves per block
    const int gemm_blocks = ((BATCH * NSEQ / 16) * (CH / 64)) / 8;  // 4096

    wmma_gemm_xWt<true ><<<gemm_blocks, 256, 0, stream>>>(x, W_attn, wbuf);
    scan1_tmp             <<<BATCH * NH, 128, 0, stream>>>(wbuf, temp, dbias, tmpb);
    head_softmax          <<<(BATCH * NSEQ) / 256, 256, 0, stream>>>(tmpb, Pi);
    scan2_y               <<<BATCH * NH, 128, 0, stream>>>(wbuf, Pi, y);
    wmma_gemm_xWt<false><<<gemm_blocks, 256, 0, stream>>>(y, W_proj, out);
}